// GCNModel_53523882443689
// MI455X (gfx1250) — compile-verified
//
#include <hip/hip_runtime.h>

#define DCH 128
#define BN_EPS 1e-5f

typedef float v2f __attribute__((ext_vector_type(2)));
typedef float v8f __attribute__((ext_vector_type(8)));

// ---------------------------------------------------------------------------
// Dense GEMM: H[n, :] = X[n, :] @ W   (W is DCH x DCH row-major, W[k][j])
// V_WMMA_F32_16X16X4_F32, wave32. One wave -> 16 rows x 128 cols.
//
// LDS holds W in a "paired + skewed" layout so a B fragment is one aligned
// ds_load_b64 into an even VGPR pair (no repack moves), and the two
// half-waves of a b64 read hit disjoint 32-bank sets (conflict-free):
//   word(rp, n, p) = rp*256 + ((2*n + p + 32*(rp&1)) & 255)
// where rp = k/2 (row pair), p = k&1.
// ---------------------------------------------------------------------------
__global__ __launch_bounds__(256) void gemm_wmma_f32(const float* __restrict__ X,
                                                     const float* __restrict__ W,
                                                     float* __restrict__ H,
                                                     int nrows) {
    __shared__ float sWp[DCH * DCH];  // 64 KB
    {
        const int t = threadIdx.x;
#pragma unroll
        for (int i = 0; i < 8; ++i) {        // 2048 chunks / 256 threads
            const int c  = t + i * 256;      // chunk id
            const int rp = c >> 5;           // row pair 0..63
            const int n4 = (c & 31) * 4;     // col base 0..124
            const float4 a4 = *(const float4*)(W + (size_t)(2 * rp) * DCH + n4);
            const float4 b4 = *(const float4*)(W + (size_t)(2 * rp + 1) * DCH + n4);
            // interleave rows k=2rp and k=2rp+1: {a.x,b.x,a.y,b.y,a.z,b.z,a.w,b.w}
            const int base = (n4 * 2 + ((rp & 1) << 5)) & 255;  // multiple of 8
            float* dstp = &sWp[rp * 256 + base];
            *(float4*)(dstp + 0) = make_float4(a4.x, b4.x, a4.y, b4.y);
            *(float4*)(dstp + 4) = make_float4(a4.z, b4.z, a4.w, b4.w);
        }
    }
    __syncthreads();

    const int lane = threadIdx.x & 31;
    const int wave = threadIdx.x >> 5;
    const int lrow = lane & 15;   // 0..15
    const int lhi  = lane >> 4;   // 0 or 1
    const int m0   = blockIdx.x * 128 + wave * 16;

    // A-fragment row for this lane (clamped so ragged tail stays in-bounds;
    // EXEC must remain all-ones for WMMA)
    int mA = m0 + lrow;
    if (mA >= nrows) mA = nrows - 1;
    const float* __restrict__ Xrow = X + (size_t)mA * DCH;

    const int c0 = 2 * lrow + 32 * lhi;  // per-lane column/skew constant

    v8f acc[8] = {};  // 8 tiles of 16x16 -> 16 rows x 128 cols

#pragma unroll 2
    for (int k0 = 0; k0 < DCH; k0 += 4) {
        const int rp = (k0 >> 1) + lhi;            // rp & 1 == lhi (k0 % 4 == 0)
        const float2 af = *(const float2*)(Xrow + k0 + 2 * lhi);  // b64 load
        v2f a;
        a.x = af.x;  // VGPR0: K = k0 + 2*lhi
        a.y = af.y;  // VGPR1: K = k0 + 2*lhi + 1
        const float* __restrict__ wrow = &sWp[rp * 256];
#pragma unroll
        for (int n = 0; n < 8; ++n) {
            const v2f b = *(const v2f*)(wrow + ((n * 32 + c0) & 255));  // ds_load_b64
            acc[n] = __builtin_amdgcn_wmma_f32_16x16x4_f32(
                false, a, false, b, (short)0, acc[n], false, false);
        }
    }

    // C/D layout: lane -> col = lane%16, VGPR v -> row = 8*(lane/16) + v
    float* __restrict__ Hbase = H + (size_t)(m0 + 8 * lhi) * DCH + lrow;
    if (m0 + 16 <= nrows) {
        // uniform fast path: whole 16-row strip in bounds, no exec churn
#pragma unroll
        for (int n = 0; n < 8; ++n)
#pragma unroll
            for (int r = 0; r < 8; ++r)
                Hbase[(size_t)r * DCH + n * 16] = acc[n][r];
    } else {
#pragma unroll
        for (int n = 0; n < 8; ++n)
#pragma unroll
            for (int r = 0; r < 8; ++r) {
                const int row = m0 + 8 * lhi + r;
                if (row < nrows) H[(size_t)row * DCH + n * 16 + lrow] = acc[n][r];
            }
    }
}

// ---------------------------------------------------------------------------
// Degree / normalization precompute
// ---------------------------------------------------------------------------
__global__ void zero_i32(int* __restrict__ p, int n) {
    int i = blockIdx.x * blockDim.x + threadIdx.x;
    if (i < n) p[i] = 0;
}

__global__ void count_deg(const int* __restrict__ dst, int* __restrict__ deg, int ne) {
    int e = blockIdx.x * blockDim.x + threadIdx.x;
    if (e < ne) atomicAdd(&deg[dst[e]], 1);
}

// in-place: reads int count, writes float rsqrt(deg+1) to same slot
__global__ void make_dinv(int* __restrict__ deg_io, int n) {
    int i = blockIdx.x * blockDim.x + threadIdx.x;
    if (i < n) {
        float d = (float)deg_io[i] + 1.0f;
        ((float*)deg_io)[i] = rsqrtf(d);
    }
}

// ---------------------------------------------------------------------------
// agg[n,:] = h[n,:] * dinv[n]^2   (self-loop term; also initializes agg)
// ---------------------------------------------------------------------------
__global__ void self_loop_init(const float* __restrict__ h,
                               const float* __restrict__ dinv,
                               float* __restrict__ agg, int nrows) {
    int i = blockIdx.x * blockDim.x + threadIdx.x;  // float4 index
    const int total = nrows * (DCH / 4);
    if (i >= total) return;
    int row = i / (DCH / 4);
    float s = dinv[row];
    s = s * s;
    float4 hv = ((const float4*)h)[i];
    float4 o;
    o.x = hv.x * s; o.y = hv.y * s; o.z = hv.z * s; o.w = hv.w * s;
    ((float4*)agg)[i] = o;
}

// ---------------------------------------------------------------------------
// Edge scatter: agg[dst,:] += h[src,:] * dinv[src]*dinv[dst]
// One wave32 per edge; lane -> 4 channels (float4) -> 128 channels/edge.
// unsafeAtomicAdd lowers to hardware global_atomic_add_f32.
// ---------------------------------------------------------------------------
__global__ __launch_bounds__(256) void edge_scatter(const float* __restrict__ h,
                                                    const int* __restrict__ src,
                                                    const int* __restrict__ dst,
                                                    const float* __restrict__ dinv,
                                                    float* __restrict__ agg, int ne) {
    const int lane = threadIdx.x & 31;
    const int wave = threadIdx.x >> 5;
    const int e = blockIdx.x * 8 + wave;
    if (e >= ne) return;
    const int s = src[e];
    const int d = dst[e];
    const float c = dinv[s] * dinv[d];
    const float4 hv = ((const float4*)(h + (size_t)s * DCH))[lane];
    float* arow = agg + (size_t)d * DCH + 4 * lane;
    unsafeAtomicAdd(arow + 0, hv.x * c);
    unsafeAtomicAdd(arow + 1, hv.y * c);
    unsafeAtomicAdd(arow + 2, hv.z * c);
    unsafeAtomicAdd(arow + 3, hv.w * c);
}

// ---------------------------------------------------------------------------
// Epilogues
// ---------------------------------------------------------------------------
__global__ void bn_relu_epilogue(const float* __restrict__ agg,
                                 const float* __restrict__ bias,
                                 const float* __restrict__ g,
                                 const float* __restrict__ be,
                                 const float* __restrict__ m,
                                 const float* __restrict__ v,
                                 float* __restrict__ out, int total) {
    int i = blockIdx.x * blockDim.x + threadIdx.x;
    if (i >= total) return;
    const int c = i & (DCH - 1);
    float x = agg[i] + bias[c];
    x = (x - m[c]) * rsqrtf(v[c] + BN_EPS) * g[c] + be[c];
    out[i] = fmaxf(x, 0.0f);
}

__global__ void residual_epilogue(const float* __restrict__ agg,
                                  const float* __restrict__ bias,
                                  const float* __restrict__ xres,
                                  float* __restrict__ out, int total) {
    int i = blockIdx.x * blockDim.x + threadIdx.x;
    if (i >= total) return;
    const int c = i & (DCH - 1);
    out[i] = agg[i] + bias[c] + xres[i];
}

// ---------------------------------------------------------------------------
// Launch
// ---------------------------------------------------------------------------
static void run_gcn_layer(const float* xin, const float* W,
                          const int* src, const int* dst, const float* dinv,
                          float* h, float* agg, int N, int E, hipStream_t stream) {
    gemm_wmma_f32<<<(N + 127) / 128, 256, 0, stream>>>(xin, W, h, N);
    const int ndv4 = N * (DCH / 4);
    self_loop_init<<<(ndv4 + 255) / 256, 256, 0, stream>>>(h, dinv, agg, N);
    edge_scatter<<<(E + 7) / 8, 256, 0, stream>>>(h, src, dst, dinv, agg, E);
}

extern "C" void kernel_launch(void* const* d_in, const int* in_sizes, int n_in,
                              void* d_out, int out_size, void* d_ws, size_t ws_size,
                              hipStream_t stream) {
    const float* x   = (const float*)d_in[0];
    const float* W1  = (const float*)d_in[1];
    const float* b1  = (const float*)d_in[2];
    const float* g1  = (const float*)d_in[3];
    const float* be1 = (const float*)d_in[4];
    const float* m1  = (const float*)d_in[5];
    const float* v1  = (const float*)d_in[6];
    const float* W2  = (const float*)d_in[7];
    const float* b2  = (const float*)d_in[8];
    const float* g2  = (const float*)d_in[9];
    const float* be2 = (const float*)d_in[10];
    const float* m2  = (const float*)d_in[11];
    const float* v2  = (const float*)d_in[12];
    const float* W3  = (const float*)d_in[13];
    const float* b3  = (const float*)d_in[14];
    const int*   src = (const int*)d_in[15];
    const int*   dst = (const int*)d_in[16];
    (void)n_in; (void)ws_size;

    const int N = in_sizes[0] / DCH;
    const int E = in_sizes[15];
    const size_t ND = (size_t)N * DCH;
    const int total = N * DCH;

    float* ws = (float*)d_ws;
    float* dinv = ws;                                     // N floats (int deg first)
    size_t off = ((size_t)N + 1023) & ~(size_t)1023;
    float* h   = ws + off;                                // N*D
    float* y0  = h + ND;                                  // N*D
    float* y1  = y0 + ND;                                 // N*D
    float* out = (float*)d_out;

    // --- degree -> dinv = rsqrt(deg+1) ---
    zero_i32<<<(N + 255) / 256, 256, 0, stream>>>((int*)dinv, N);
    count_deg<<<(E + 255) / 256, 256, 0, stream>>>(dst, (int*)dinv, E);
    make_dinv<<<(N + 255) / 256, 256, 0, stream>>>((int*)dinv, N);

    // --- layer 1: x -> y0 ---
    run_gcn_layer(x, W1, src, dst, dinv, h, y0, N, E, stream);
    bn_relu_epilogue<<<(total + 255) / 256, 256, 0, stream>>>(y0, b1, g1, be1, m1, v1, y0, total);

    // --- layer 2: y0 -> y1 ---
    run_gcn_layer(y0, W2, src, dst, dinv, h, y1, N, E, stream);
    bn_relu_epilogue<<<(total + 255) / 256, 256, 0, stream>>>(y1, b2, g2, be2, m2, v2, y1, total);

    // --- layer 3: y1 -> out (+ residual x) ---
    run_gcn_layer(y1, W3, src, dst, dinv, h, y0, N, E, stream);
    residual_epilogue<<<(total + 255) / 256, 256, 0, stream>>>(y0, b3, x, out, total);
    (void)out_size;
}